// GraphConvTest_22084721836306
// MI455X (gfx1250) — compile-verified
//
#include <hip/hip_runtime.h>
#include <hip/hip_bf16.h>
#include <math.h>

// CDNA5 (gfx1250) fp32 WMMA: D(16x16) = A(16x4) * B(4x16) + C
typedef __attribute__((ext_vector_type(2))) float v2f;
typedef __attribute__((ext_vector_type(4))) float v4f;
typedef __attribute__((ext_vector_type(8))) float v8f;

__device__ __forceinline__ v8f wmma4(v2f a, v2f b, v8f c) {
    return __builtin_amdgcn_wmma_f32_16x16x4_f32(false, a, false, b, (short)0, c, false, false);
}

__device__ __forceinline__ v8f zero8() {
    v8f z;
#pragma unroll
    for (int i = 0; i < 8; ++i) z[i] = 0.0f;
    return z;
}

#define BATCH 16
#define NNODE 256
#define FDIM  33
#define HDIM  11
#define FP    48   // padded feature width
#define KOUT  364  // 363 padded to multiple of 4

// float offsets inside d_ws
#define OFF_WT   0u                      // 36x48 padded test_W
#define OFF_WO   1728u                   // 364x48 padded out_linear_W
#define OFF_WA   19200u                  // 2 x 36x48 padded adj_linear_W halves
#define OFF_W2   22656u                  // 44x16 padded adj_out_linear_W
#define OFF_SUP  23360u                  // supP   (4096 x 48)
#define OFF_O2P  219968u                 // out2P  (4096 x 48)
#define OFF_P    416576u                 // P      (4096 x 48)
#define OFF_Q    613184u                 // Q      (4096 x 48)
#define OFF_MSG  809792u                 // msgt2  (4096 x 364)
#define PACK_N   27456

// ---------------------------------------------------------------------------
// K0: pack weights zero-padded into ws; zero msgt2 pad column (k=363).
// ---------------------------------------------------------------------------
__global__ __launch_bounds__(256) void k_pack(const float* __restrict__ tW,
                                              const float* __restrict__ oW,
                                              const float* __restrict__ aW,
                                              const float* __restrict__ a2W,
                                              float* __restrict__ ws) {
    int idx = blockIdx.x * 256 + threadIdx.x;
    if (idx < 1728) {                       // test_W (33x33) -> 36x48
        int k = idx / FP, n = idx % FP;
        ws[OFF_WT + idx] = (k < FDIM && n < FDIM) ? tW[k * FDIM + n] : 0.0f;
    } else if (idx < 19200) {               // out_linear_W (363x33) -> 364x48
        int r = idx - 1728, k = r / FP, n = r % FP;
        ws[OFF_WO + r] = (k < 363 && n < FDIM) ? oW[k * FDIM + n] : 0.0f;
    } else if (idx < 22656) {               // adj_linear_W (66x33) -> 2 x 36x48
        int r = idx - 19200, z = r / 1728, r2 = r % 1728;
        int k = r2 / FP, n = r2 % FP;
        ws[OFF_WA + r] = (k < FDIM && n < FDIM) ? aW[(z * FDIM + k) * FDIM + n] : 0.0f;
    } else if (idx < 23360) {               // adj_out_linear_W (44x11) -> 44x16
        int r = idx - 22656, k = r / 16, h = r % 16;
        ws[OFF_W2 + r] = (h < HDIM) ? a2W[k * HDIM + h] : 0.0f;
    } else if (idx < PACK_N) {              // zero msgt2[row][363]
        int row = idx - 23360;
        ws[OFF_MSG + (size_t)row * KOUT + 363] = 0.0f;
    }
}

// ---------------------------------------------------------------------------
// K1: supP[row, 0:48] = node_feature[row,:] @ WtP
// ---------------------------------------------------------------------------
__global__ __launch_bounds__(32) void k_support(const float* __restrict__ nf,
                                                const float* __restrict__ WtP,
                                                float* __restrict__ supP) {
    const int lane = threadIdx.x, half = lane >> 4, m = lane & 15;
    const int row0 = blockIdx.x * 16, row = row0 + m;
    const int n = blockIdx.y * 16 + m;
    v8f acc = zero8();
#pragma unroll
    for (int k0 = 0; k0 < 32; k0 += 4) {
        const int ka = k0 + half * 2;
        v2f a, b;
        a.x = nf[row * FDIM + ka];
        a.y = nf[row * FDIM + ka + 1];
        b.x = WtP[ka * FP + n];
        b.y = WtP[(ka + 1) * FP + n];
        acc = wmma4(a, b, acc);
    }
    {   // boundary: only k=32 contributes (W pad rows 33..35 are zero)
        const int ka = 32 + half * 2;
        v2f a, b;
        a.x = (half == 0) ? nf[row * FDIM + 32] : 0.0f;
        a.y = 0.0f;
        b.x = WtP[ka * FP + n];
        b.y = WtP[(ka + 1) * FP + n];
        acc = wmma4(a, b, acc);
    }
#pragma unroll
    for (int r = 0; r < 8; ++r)
        supP[(size_t)(row0 + half * 8 + r) * FP + n] = acc[r];
}

// ---------------------------------------------------------------------------
// K2: msgt2[b,i, f*11+h] = tanh( sum_j lcq[b,i,j,h]*sup[b,j,f] + sup[b,i,f] )
// 8 waves/block share the full 256x48 supP[b] panel in LDS; each wave does one
// i-tile with 3 f-tile accumulators (A loaded once, B from LDS).
// grid: (2 i-groups, 176 b*h), block: 256
// ---------------------------------------------------------------------------
__global__ __launch_bounds__(256) void k_message(const float* __restrict__ lcq,
                                                 const float* __restrict__ supP,
                                                 float* __restrict__ msgt2) {
    __shared__ float lds[NNODE * FP]; // 48 KB
    const int t = threadIdx.x, lane = t & 31, w = t >> 5;
    const int half = lane >> 4, m = lane & 15;
    const int bh = blockIdx.y, b = bh / HDIM, h = bh - b * HDIM;
    const float* supB = supP + (size_t)b * NNODE * FP;
#pragma unroll
    for (int r = 0; r < 12; ++r) {
        const int off = r * 1024 + t * 4;
        *(v4f*)&lds[off] = *(const v4f*)&supB[off];
    }
    __syncthreads();

    const int i0 = (blockIdx.x * 8 + w) * 16;
    const float* lcq_b = lcq + (size_t)b * NNODE * NNODE * HDIM;
    v8f acc0 = zero8(), acc1 = zero8(), acc2 = zero8();
#pragma unroll 2
    for (int j0 = 0; j0 < NNODE; j0 += 4) {
        const int ka = j0 + half * 2;
        v2f a, b0, b1, b2;
        const size_t base = ((size_t)(i0 + m) * NNODE + ka) * HDIM + h;
        a.x = lcq_b[base];
        a.y = lcq_b[base + HDIM];
        b0.x = lds[ka * FP + m];       b0.y = lds[(ka + 1) * FP + m];
        b1.x = lds[ka * FP + 16 + m];  b1.y = lds[(ka + 1) * FP + 16 + m];
        b2.x = lds[ka * FP + 32 + m];  b2.y = lds[(ka + 1) * FP + 32 + m];
        acc0 = wmma4(a, b0, acc0);
        acc1 = wmma4(a, b1, acc1);
        acc2 = wmma4(a, b2, acc2);
    }
#pragma unroll
    for (int ft = 0; ft < 3; ++ft) {
        const int n = ft * 16 + m;
        const v8f acc = (ft == 0) ? acc0 : (ft == 1) ? acc1 : acc2;
        if (n < FDIM) {
#pragma unroll
            for (int r = 0; r < 8; ++r) {
                const int i = i0 + half * 8 + r;
                const float v = acc[r] + lds[i * FP + n];
                msgt2[((size_t)b * NNODE + i) * KOUT + n * HDIM + h] = tanhf(v);
            }
        }
    }
}

// ---------------------------------------------------------------------------
// K3: out2[row,u] = tanh( msgt2[row,:] @ WoP ), K=364; 3 u-tiles per wave so
// one aligned b64 A-load feeds 3 WMMAs. grid: (256 row-tiles)
// ---------------------------------------------------------------------------
__global__ __launch_bounds__(32) void k_outlin(const float* __restrict__ msgt2,
                                               const float* __restrict__ WoP,
                                               float* __restrict__ out2,
                                               float* __restrict__ out2P) {
    const int lane = threadIdx.x, half = lane >> 4, m = lane & 15;
    const int row0 = blockIdx.x * 16, row = row0 + m;
    v8f acc0 = zero8(), acc1 = zero8(), acc2 = zero8();
    for (int k0 = 0; k0 < KOUT; k0 += 4) {
        const int ka = k0 + half * 2;
        const v2f a = *(const v2f*)&msgt2[(size_t)row * KOUT + ka];
        v2f b0, b1, b2;
        b0.x = WoP[ka * FP + m];       b0.y = WoP[(ka + 1) * FP + m];
        b1.x = WoP[ka * FP + 16 + m];  b1.y = WoP[(ka + 1) * FP + 16 + m];
        b2.x = WoP[ka * FP + 32 + m];  b2.y = WoP[(ka + 1) * FP + 32 + m];
        acc0 = wmma4(a, b0, acc0);
        acc1 = wmma4(a, b1, acc1);
        acc2 = wmma4(a, b2, acc2);
    }
#pragma unroll
    for (int ft = 0; ft < 3; ++ft) {
        const int u = ft * 16 + m;
        const v8f acc = (ft == 0) ? acc0 : (ft == 1) ? acc1 : acc2;
#pragma unroll
        for (int r = 0; r < 8; ++r) {
            const float tv = tanhf(acc[r]);
            const int rr = row0 + half * 8 + r;
            out2P[(size_t)rr * FP + u] = tv;
            if (u < FDIM) out2[rr * FDIM + u] = tv;
        }
    }
}

// ---------------------------------------------------------------------------
// K4: P = out2P @ WaP[0], Q = out2P @ WaP[1]. grid: (256, 3, 2)
// ---------------------------------------------------------------------------
__global__ __launch_bounds__(32) void k_pq(const float* __restrict__ out2P,
                                           const float* __restrict__ WaP,
                                           float* __restrict__ P,
                                           float* __restrict__ Q) {
    const int lane = threadIdx.x, half = lane >> 4, m = lane & 15;
    const int row0 = blockIdx.x * 16, row = row0 + m;
    const int u = blockIdx.y * 16 + m;
    const float* Wp = WaP + (size_t)blockIdx.z * 36 * FP;
    float* dst = blockIdx.z ? Q : P;
    v8f acc = zero8();
#pragma unroll
    for (int k0 = 0; k0 < 36; k0 += 4) {
        const int ka = k0 + half * 2;
        const v2f a = *(const v2f*)&out2P[(size_t)row * FP + ka];
        v2f b;
        b.x = Wp[ka * FP + u];
        b.y = Wp[(ka + 1) * FP + u];
        acc = wmma4(a, b, acc);
    }
#pragma unroll
    for (int r = 0; r < 8; ++r)
        dst[(size_t)(row0 + half * 8 + r) * FP + u] = acc[r];
}

// ---------------------------------------------------------------------------
// K5: fused edge kernel. Per wave: fixed (b, j-tile), loop over 32 i's.
// B-frags (W2) and P-frags held in registers across the i-loop; Q[b] panel in
// LDS (broadcast reads). Inner i-iteration = 11 WMMAs (K=44 exact).
// grid: (2 j-groups, 8 i-chunks, 16 b), block: 256 (8 waves)
// ---------------------------------------------------------------------------
__global__ __launch_bounds__(256) void k_edge(const float* __restrict__ P,
                                              const float* __restrict__ Q,
                                              const float* __restrict__ lcq,
                                              const float* __restrict__ W2P,
                                              float* __restrict__ adj) {
    __shared__ float qlds[NNODE * FP]; // 48 KB
    const int t = threadIdx.x, lane = t & 31, w = t >> 5;
    const int half = lane >> 4, m = lane & 15;
    const int b = blockIdx.z;
    const int i0 = blockIdx.y * 32;
    const int j0 = (blockIdx.x * 8 + w) * 16;

    const float* Qb = Q + (size_t)b * NNODE * FP;
#pragma unroll
    for (int r = 0; r < 12; ++r) {
        const int off = r * 1024 + t * 4;
        *(v4f*)&qlds[off] = *(const v4f*)&Qb[off];
    }
    __syncthreads();

    // Preload B fragments (W2P, 44x16) and P fragments (row j0+m) once.
    v2f bf[11];
#pragma unroll
    for (int s = 0; s < 11; ++s) {
        const int ka = s * 4 + half * 2;
        bf[s].x = W2P[ka * 16 + m];
        bf[s].y = W2P[(ka + 1) * 16 + m];
    }
    const float* Prow = P + ((size_t)b * NNODE + j0 + m) * FP;
    float pa[18];
#pragma unroll
    for (int s = 0; s < 9; ++s) {
        const int ka = s * 4 + half * 2;
        pa[2 * s]     = Prow[ka];
        pa[2 * s + 1] = Prow[ka + 1];
    }

    const float* Lbase = lcq + (size_t)b * NNODE * NNODE * HDIM;
    for (int ii = 0; ii < 32; ++ii) {
        const int i = i0 + ii;
        const float* Lb = Lbase + (size_t)i * NNODE * HDIM + (size_t)(j0 + m) * HDIM;
        v8f acc = zero8();
#pragma unroll
        for (int s = 0; s < 8; ++s) {          // k = 0..31: relu(P+Q)
            const int ka = s * 4 + half * 2;
            v2f a;
            a.x = fmaxf(pa[2 * s]     + qlds[i * FP + ka],     0.0f);
            a.y = fmaxf(pa[2 * s + 1] + qlds[i * FP + ka + 1], 0.0f);
            acc = wmma4(a, bf[s], acc);
        }
        {   // s=8: k=32..35 mixed (k=32 -> P+Q; k=33..35 -> lcq[0..2])
            v2f a;
            const float Lx = Lb[(half == 0) ? 0 : 1];
            const float Px = fmaxf(pa[16] + qlds[i * FP + 32], 0.0f);
            a.x = (half == 0) ? Px : Lx;
            a.y = Lb[(half == 0) ? 0 : 2];
            acc = wmma4(a, bf[8], acc);
        }
#pragma unroll
        for (int s = 9; s < 11; ++s) {         // k = 36..43 -> lcq[3..10]
            const int ka = s * 4 + half * 2;
            v2f a;
            a.x = Lb[ka - 33];
            a.y = Lb[ka - 32];
            acc = wmma4(a, bf[s], acc);
        }
        if (m < HDIM) {
#pragma unroll
            for (int r = 0; r < 8; ++r) {
                const int j = j0 + half * 8 + r;
                const size_t idx = (((size_t)b * NNODE + i) * NNODE + j) * HDIM + m;
                adj[idx] = fmaxf(acc[r], 0.0f) + lcq[idx];
            }
        }
    }
}

// ---------------------------------------------------------------------------
extern "C" void kernel_launch(void* const* d_in, const int* in_sizes, int n_in,
                              void* d_out, int out_size, void* d_ws, size_t ws_size,
                              hipStream_t stream) {
    (void)in_sizes; (void)n_in; (void)out_size; (void)ws_size;
    const float* node_feature = (const float*)d_in[0]; // (16,256,33)
    const float* lcq_adj      = (const float*)d_in[1]; // (16,256,256,11)
    const float* test_W       = (const float*)d_in[2]; // (33,33)
    const float* out_linear_W = (const float*)d_in[3]; // (363,33)
    const float* adj_linear_W = (const float*)d_in[4]; // (66,33)
    const float* adj_out_W    = (const float*)d_in[5]; // (44,11)

    float* out2 = (float*)d_out;                                // (16,256,33)
    float* adj  = (float*)d_out + (size_t)BATCH * NNODE * FDIM; // (16,256,256,11)

    float* ws = (float*)d_ws;
    const dim3 wave(32);

    k_pack<<<dim3((PACK_N + 255) / 256), dim3(256), 0, stream>>>(
        test_W, out_linear_W, adj_linear_W, adj_out_W, ws);

    k_support<<<dim3(BATCH * NNODE / 16, 3), wave, 0, stream>>>(
        node_feature, ws + OFF_WT, ws + OFF_SUP);

    k_message<<<dim3(2, BATCH * HDIM), dim3(256), 0, stream>>>(
        lcq_adj, ws + OFF_SUP, ws + OFF_MSG);

    k_outlin<<<dim3(BATCH * NNODE / 16), wave, 0, stream>>>(
        ws + OFF_MSG, ws + OFF_WO, out2, ws + OFF_O2P);

    k_pq<<<dim3(BATCH * NNODE / 16, 3, 2), wave, 0, stream>>>(
        ws + OFF_O2P, ws + OFF_WA, ws + OFF_P, ws + OFF_Q);

    k_edge<<<dim3(2, 8, BATCH), dim3(256), 0, stream>>>(
        ws + OFF_P, ws + OFF_Q, lcq_adj, ws + OFF_W2, adj);
}